// RealTimeLinearRTUsLayer_53712861003855
// MI455X (gfx1250) — compile-verified
//
#include <hip/hip_runtime.h>
#include <hip/hip_bf16.h>

// ---------------------------------------------------------------------------
// RealTimeLinearRTUs layer step, MI455X (gfx1250).
// BT=128, D=256, H=512.
//
// Roofline: dominated by streaming 4x(BT,D,H) f32 reads + 4x writes
// (~512 MiB => ~23 us @ 23.3 TB/s). GEMMs are ~67 MFLOP (noise) but are done
// with V_WMMA_F32_16X16X4_F32 to use the CDNA5 matrix path at full f32
// fidelity. Streaming tensors use non-temporal b128 loads/stores so they do
// not thrash the 192 MB L2.
// ---------------------------------------------------------------------------

#define BT 128
#define DD 256
#define HH 512

typedef float v2f __attribute__((ext_vector_type(2)));
typedef float v4f __attribute__((ext_vector_type(4)));
typedef float v8f __attribute__((ext_vector_type(8)));

// ---------------------------------------------------------------------------
// Kernel 0: per-H coefficients (8 vectors of length H) -> ws coef block.
// coef rows: 0:A 1:B 2:g 3:dA_dnu 4:dB_dnu 5:dg_dnu 6:dA_dth 7:dB_dth
// ---------------------------------------------------------------------------
__global__ __launch_bounds__(256) void rtu_coeffs(
    const float* __restrict__ nu_log, const float* __restrict__ theta_log,
    float* __restrict__ coef)
{
    int h = blockIdx.x * blockDim.x + threadIdx.x;
    if (h >= HH) return;
    float exp_nu = expf(nu_log[h]);
    float r      = expf(-exp_nu);
    float theta  = expf(theta_log[h]);
    float g      = sqrtf(1.0f - r * r);
    float A      = r * cosf(theta);
    float B      = r * sinf(theta);
    coef[0 * HH + h] = A;
    coef[1 * HH + h] = B;
    coef[2 * HH + h] = g;
    coef[3 * HH + h] = -A * exp_nu;              // dA_dnu
    coef[4 * HH + h] = -B * exp_nu;              // dB_dnu
    coef[5 * HH + h] = (r * r) * exp_nu / g;     // dg_dnu
    coef[6 * HH + h] = -B * theta;               // dA_dth
    coef[7 * HH + h] =  A * theta;               // dB_dth
}

// ---------------------------------------------------------------------------
// Kernel 1: xw1 = x_t @ w1, xw2 = x_t @ w2 via V_WMMA_F32_16X16X4_F32.
// One wave computes one 16x16 tile of one output (8x32 tiles x 2 matrices).
// Fragment layouts per CDNA5 ISA 7.12.2:
//   A (16x4 f32):  lanes 0-15 rows M=0..15 with K={0,1}; lanes 16-31 K={2,3}.
//   B (4x16 f32):  lanes 0-15 cols N=0..15 with K={0,1}; lanes 16-31 K={2,3}.
//   C/D (16x16):   VGPR r holds M=r (lanes 0-15) / M=8+r (lanes 16-31), N=lane&15.
// ---------------------------------------------------------------------------
__global__ __launch_bounds__(128) void rtu_gemm_wmma(
    const float* __restrict__ x,
    const float* __restrict__ w1, const float* __restrict__ w2,
    float* __restrict__ xw1, float* __restrict__ xw2)
{
    const int wave = threadIdx.x >> 5;
    const int lane = threadIdx.x & 31;
    int tile = blockIdx.x * 4 + wave;            // 0..511
    const int mat = tile >> 8;                   // 0 -> w1/xw1, 1 -> w2/xw2
    tile &= 255;
    const int tm = tile >> 5;                    // 0..7  (M tile)
    const int tn = tile & 31;                    // 0..31 (N tile)

    const float* __restrict__ w   = mat ? w2  : w1;
    float*       __restrict__ dst = mat ? xw2 : xw1;

    const int arow = lane & 15;
    const int koff = (lane >> 4) << 1;           // 0 or 2
    const int bcol = lane & 15;

    const float* aptr = x + (size_t)(tm * 16 + arow) * DD + koff;
    const float* bptr = w + (size_t)koff * HH + tn * 16 + bcol;

    v8f acc = {};
    #pragma unroll 4
    for (int k = 0; k < DD; k += 4) {
        v2f a, b;
        a.x = aptr[k];
        a.y = aptr[k + 1];
        b.x = bptr[(size_t)k * HH];
        b.y = bptr[(size_t)(k + 1) * HH];
        // 8 args: (neg_a, A, neg_b, B, c_mod, C, reuse_a, reuse_b)
        acc = __builtin_amdgcn_wmma_f32_16x16x4_f32(
            false, a, false, b, (short)0, acc, false, false);
    }

    const int crow0 = (lane >> 4) * 8;
    const int ccol  = lane & 15;
    #pragma unroll
    for (int r = 0; r < 8; ++r)
        dst[(size_t)(tm * 16 + crow0 + r) * HH + tn * 16 + ccol] = acc[r];
}

// ---------------------------------------------------------------------------
// Kernel 2: fused small elementwise: c1,c2,n1,n2,t1,t2 and h_t = relu(cat).
// (BT*H = 65536 threads; tiny vs the streaming kernel.)
// ---------------------------------------------------------------------------
__global__ __launch_bounds__(256) void rtu_small(
    const float* __restrict__ h1,   const float* __restrict__ h2,
    const float* __restrict__ mnu1, const float* __restrict__ mnu2,
    const float* __restrict__ mth1, const float* __restrict__ mth2,
    const float* __restrict__ xw1,  const float* __restrict__ xw2,
    const float* __restrict__ coef,
    float* __restrict__ ht,
    float* __restrict__ c1o, float* __restrict__ c2o,
    float* __restrict__ n1o, float* __restrict__ n2o,
    float* __restrict__ t1o, float* __restrict__ t2o)
{
    int i = blockIdx.x * 256 + threadIdx.x;      // < BT*HH
    int h = i & (HH - 1);
    int b = i >> 9;

    float A   = coef[0 * HH + h];
    float B   = coef[1 * HH + h];
    float g   = coef[2 * HH + h];
    float dAn = coef[3 * HH + h];
    float dBn = coef[4 * HH + h];
    float dgn = coef[5 * HH + h];
    float dAt = coef[6 * HH + h];
    float dBt = coef[7 * HH + h];

    float H1 = h1[i],  H2 = h2[i];
    float X1 = xw1[i], X2 = xw2[i];

    float c1 = A * H1 - B * H2 + g * X1;
    float c2 = B * H1 + A * H2 + g * X2;

    float M1 = mnu1[i], M2 = mnu2[i];
    float n1 = A * M1 - B * M2 + dAn * H1 - dBn * H2 + dgn * X1;
    float n2 = B * M1 + A * M2 + dBn * H1 + dAn * H2 + dgn * X2;

    float T1 = mth1[i], T2 = mth2[i];
    float t1 = A * T1 - B * T2 + dAt * H1 - dBt * H2;
    float t2 = B * T1 + A * T2 + dBt * H1 + dAt * H2;

    c1o[i] = c1; c2o[i] = c2;
    n1o[i] = n1; n2o[i] = n2;
    t1o[i] = t1; t2o[i] = t2;

    ht[(size_t)b * (2 * HH) + h]      = fmaxf(c1, 0.0f);
    ht[(size_t)b * (2 * HH) + HH + h] = fmaxf(c2, 0.0f);
}

// ---------------------------------------------------------------------------
// Kernel 3: THE kernel — streams 4x (BT,D,H) in, 4x out (512 MiB total).
// One float4 (b128) per thread along H; non-temporal to bypass L2 retention.
// ---------------------------------------------------------------------------
__global__ __launch_bounds__(256) void rtu_big(
    const float* __restrict__ x, const float* __restrict__ coef,
    const float* __restrict__ mw11, const float* __restrict__ mw12,
    const float* __restrict__ mw21, const float* __restrict__ mw22,
    float* __restrict__ W11, float* __restrict__ W12,
    float* __restrict__ W21, float* __restrict__ W22)
{
    size_t v = (size_t)blockIdx.x * 256 + threadIdx.x;   // vec4 index
    int    h4 = (int)(v & (HH / 4 - 1));                 // 0..127
    size_t bd = v >> 7;                                  // b*D + d

    float xval = x[bd];
    v4f A = *(const v4f*)(coef + 0 * HH + h4 * 4);
    v4f B = *(const v4f*)(coef + 1 * HH + h4 * 4);
    v4f g = *(const v4f*)(coef + 2 * HH + h4 * 4);
    v4f gx = g * xval;

    v4f m11 = __builtin_nontemporal_load((const v4f*)mw11 + v);
    v4f m12 = __builtin_nontemporal_load((const v4f*)mw12 + v);
    v4f m21 = __builtin_nontemporal_load((const v4f*)mw21 + v);
    v4f m22 = __builtin_nontemporal_load((const v4f*)mw22 + v);

    __builtin_nontemporal_store(A * m11 - B * m12 + gx, (v4f*)W11 + v);
    __builtin_nontemporal_store(B * m11 + A * m12,      (v4f*)W12 + v);
    __builtin_nontemporal_store(A * m21 - B * m22,      (v4f*)W21 + v);
    __builtin_nontemporal_store(B * m21 + A * m22 + gx, (v4f*)W22 + v);
}

// ---------------------------------------------------------------------------
extern "C" void kernel_launch(void* const* d_in, const int* in_sizes, int n_in,
                              void* d_out, int out_size, void* d_ws, size_t ws_size,
                              hipStream_t stream) {
    (void)in_sizes; (void)n_in; (void)out_size; (void)ws_size;

    const float* x_t       = (const float*)d_in[0];
    const float* h1        = (const float*)d_in[1];
    const float* h2        = (const float*)d_in[2];
    const float* mnu1      = (const float*)d_in[3];
    const float* mnu2      = (const float*)d_in[4];
    const float* mth1      = (const float*)d_in[5];
    const float* mth2      = (const float*)d_in[6];
    const float* mw11      = (const float*)d_in[7];
    const float* mw12      = (const float*)d_in[8];
    const float* mw21      = (const float*)d_in[9];
    const float* mw22      = (const float*)d_in[10];
    const float* w1        = (const float*)d_in[11];
    const float* w2        = (const float*)d_in[12];
    const float* nu_log    = (const float*)d_in[13];
    const float* theta_log = (const float*)d_in[14];

    // workspace: xw1 (BT*H) | xw2 (BT*H) | coef (8*H)
    float* ws   = (float*)d_ws;
    float* xw1  = ws;
    float* xw2  = ws + (size_t)BT * HH;
    float* coef = ws + (size_t)2 * BT * HH;

    // output layout (flat, reference tuple order)
    float* out = (float*)d_out;
    const size_t nBH = (size_t)BT * HH;          // 65536
    const size_t nW  = (size_t)BT * DD * HH;     // 16777216
    float* ht  = out;
    float* c1  = out + (size_t)BT * 2 * HH;
    float* c2  = c1 + nBH;
    float* n1  = c2 + nBH;
    float* n2  = n1 + nBH;
    float* t1  = n2 + nBH;
    float* t2  = t1 + nBH;
    float* W11 = t2 + nBH;
    float* W12 = W11 + nW;
    float* W21 = W12 + nW;
    float* W22 = W21 + nW;

    // 0) coefficients (512 lanes)
    rtu_coeffs<<<(HH + 255) / 256, 256, 0, stream>>>(nu_log, theta_log, coef);

    // 1) WMMA GEMMs: 512 tiles, 4 waves/block -> 128 blocks
    rtu_gemm_wmma<<<128, 128, 0, stream>>>(x_t, w1, w2, xw1, xw2);

    // 2) fused small elementwise (BT*H threads)
    rtu_small<<<(int)(nBH / 256), 256, 0, stream>>>(
        h1, h2, mnu1, mnu2, mth1, mth2, xw1, xw2, coef,
        ht, c1, c2, n1, n2, t1, t2);

    // 3) streaming trace update: BT*D*H/4 float4's, one per thread
    rtu_big<<<(int)(nW / 4 / 256), 256, 0, stream>>>(
        x_t, coef, mw11, mw12, mw21, mw22, W11, W12, W21, W22);
}